// GAT_49821620634001
// MI455X (gfx1250) — compile-verified
//
#include <hip/hip_runtime.h>
#include <hip/hip_bf16.h>
#include <math.h>

// ---------------------------------------------------------------------------
// GAT (3 layers + classifier) for MI455X / gfx1250, wave32, WMMA f32 path.
// ---------------------------------------------------------------------------

typedef float v2f __attribute__((ext_vector_type(2)));
typedef float v8f __attribute__((ext_vector_type(8)));

#define N_NODES 50000
#define N_EDGES 800000
#define ETOT    (N_EDGES + N_NODES)   // self-loops appended after real edges

// ---- monotone float<->uint key for atomicMax on signed floats --------------
__device__ __forceinline__ unsigned fkey(float f) {
  unsigned b = __float_as_uint(f);
  return (b & 0x80000000u) ? ~b : (b | 0x80000000u);
}
__device__ __forceinline__ float funkey(unsigned k) {
  return (k & 0x80000000u) ? __uint_as_float(k ^ 0x80000000u)
                           : __uint_as_float(~k);
}
// key(-inf) = ~0xFF800000 = 0x007FFFFF
#define KEY_NEG_INF 0x007FFFFFu

__device__ __forceinline__ float lrelu(float v) {
  return v > 0.f ? v : 0.2f * v;
}

__device__ __forceinline__ void edge_nodes(const long long* __restrict__ ei,
                                           int e, int& s, int& d) {
  if (e < N_EDGES) {
    s = (int)ei[e];
    d = (int)ei[(size_t)N_EDGES + e];
  } else {
    s = d = e - N_EDGES;   // self loop
  }
}

// ---------------------------------------------------------------------------
// Fill kernels (scratch init; harness poisons ws with 0xAA)
// ---------------------------------------------------------------------------
__global__ void fill_u32(unsigned* __restrict__ p, unsigned v, int n) {
  int i = blockIdx.x * blockDim.x + threadIdx.x;
  if (i < n) p[i] = v;
}
__global__ void fill_f32(float* __restrict__ p, float v, int n) {
  int i = blockIdx.x * blockDim.x + threadIdx.x;
  if (i < n) p[i] = v;
}

// ---------------------------------------------------------------------------
// WMMA fp32 GEMM: C[M,NC] = A[M,K] @ W[K,NC] (+bias, +activation).
// One wave computes a 16x16 tile via V_WMMA_F32_16X16X4_F32 over K/4 steps.
// blockDim.x = 128 (4 waves tiled along N). M = 50000 = 3125*16 (no row guard).
//
// Fragment mapping (ISA 7.12.2, 32-bit A 16x4 / 32-bit C 16x16):
//   A: lane<16 -> K={k0,k0+1}, lane>=16 -> K={k0+2,k0+3}, row M = m0+(lane&15)
//   B: lane<16 -> rows {k0,k0+1}, lane>=16 -> rows {k0+2,k0+3}, col = n0+(lane&15)
//   C/D: VGPR r -> M = m0 + r + 8*(lane>>4), N = n0+(lane&15)
//
// GUARD (classifier NC=40): out-of-range B lanes only feed the output column
// owned by that same lane, which is never stored -> clamp the column for the
// loads (stay in-bounds, no EXEC churn) and predicate only the final store.
// ACT: 0=none 1=relu 2=tanh
// ---------------------------------------------------------------------------
template <int K, int NC, bool GUARD, int ACT>
__global__ __launch_bounds__(128)
void gemm_wmma_f32(const float* __restrict__ A, const float* __restrict__ W,
                   const float* __restrict__ bias, float* __restrict__ C)
{
  const int wave = threadIdx.x >> 5;
  const int lane = threadIdx.x & 31;
  const int half = lane >> 4;
  const int ln   = lane & 15;
  const int m0   = blockIdx.y * 16;
  const int n0   = (blockIdx.x * 4 + wave) * 16;
  if (GUARD && n0 >= NC) return;           // wave-uniform exit

  const int  row  = m0 + ln;               // A row this lane reads
  const int  col  = n0 + ln;               // C column this lane owns
  const int  colc = GUARD ? (col < NC ? col : NC - 1) : col;  // clamped, in-bounds

  const float* __restrict__ ap = A + (size_t)row * K + 2 * half;      // 8B aligned
  const float* __restrict__ wp = W + (size_t)(2 * half) * NC + colc;

  v8f acc = {};
  #pragma unroll 16
  for (int k0 = 0; k0 < K; k0 += 4) {
    v2f a = *(const v2f*)ap;               // global_load_b64, immediate offset
    v2f b;
    b.x = wp[0];                           // global_load_b32 @ +0
    b.y = wp[NC];                          // global_load_b32 @ +NC*4 (immediate)
    acc = __builtin_amdgcn_wmma_f32_16x16x4_f32(
        /*neg_a=*/false, a, /*neg_b=*/false, b,
        /*c_mod=*/(short)0, acc, /*reuse_a=*/false, /*reuse_b=*/false);
    ap += 4;
    wp += 4 * NC;
  }

  if (!GUARD || col < NC) {
    const float bv = bias ? bias[col] : 0.f;
    float* __restrict__ cp = C + (size_t)(m0 + 8 * half) * NC + col;
    #pragma unroll
    for (int r = 0; r < 8; ++r) {
      float v = acc[r] + bv;
      if (ACT == 1) v = fmaxf(v, 0.f);
      else if (ACT == 2) v = tanhf(v);
      cp[(size_t)r * NC] = v;              // immediate offsets, stride NC
    }
  }
}

// ---------------------------------------------------------------------------
// Per-(node, head) attention logits: as[n,h] = <h[n,h,:], a_src[h,:]>, same ad.
// One wave32 per (node, head), shuffle reduction.
// ---------------------------------------------------------------------------
__global__ __launch_bounds__(256)
void alpha_kernel(const float* __restrict__ hmat,
                  const float* __restrict__ a_src,
                  const float* __restrict__ a_dst,
                  float* __restrict__ as, float* __restrict__ ad,
                  int H, int C)
{
  const int wid  = (int)((blockIdx.x * blockDim.x + threadIdx.x) >> 5);
  const int lane = threadIdx.x & 31;
  if (wid >= N_NODES * H) return;
  const int n = wid / H, hh = wid % H;

  const float* __restrict__ hp  = hmat + (size_t)n * H * C + (size_t)hh * C;
  const float* __restrict__ asp = a_src + hh * C;
  const float* __restrict__ adp = a_dst + hh * C;

  float s = 0.f, d = 0.f;
  for (int c = lane; c < C; c += 32) {
    const float v = hp[c];
    s += v * asp[c];
    d += v * adp[c];
  }
  #pragma unroll
  for (int off = 16; off >= 1; off >>= 1) {
    s += __shfl_down(s, off, 32);
    d += __shfl_down(d, off, 32);
  }
  if (lane == 0) { as[wid] = s; ad[wid] = d; }
}

// ---------------------------------------------------------------------------
// Edge pass 1: segment max of leaky_relu(as[src]+ad[dst]) per (dst, head).
// ---------------------------------------------------------------------------
__global__ __launch_bounds__(256)
void edge_max_kernel(const long long* __restrict__ ei,
                     const float* __restrict__ as, const float* __restrict__ ad,
                     unsigned* __restrict__ smax, int H)
{
  const int e = blockIdx.x * blockDim.x + threadIdx.x;
  if (e >= ETOT) return;
  int s, d; edge_nodes(ei, e, s, d);
  for (int hh = 0; hh < H; ++hh) {
    const float v = lrelu(as[s * H + hh] + ad[d * H + hh]);
    atomicMax(&smax[d * H + hh], fkey(v));
  }
}

// ---------------------------------------------------------------------------
// Edge pass 2: ex = exp(e - max[dst]); store per-edge ex; segment-sum denom.
// ---------------------------------------------------------------------------
__global__ __launch_bounds__(256)
void edge_expsum_kernel(const long long* __restrict__ ei,
                        const float* __restrict__ as, const float* __restrict__ ad,
                        const unsigned* __restrict__ smax,
                        float* __restrict__ ssum, float* __restrict__ eex, int H)
{
  const int e = blockIdx.x * blockDim.x + threadIdx.x;
  if (e >= ETOT) return;
  int s, d; edge_nodes(ei, e, s, d);
  for (int hh = 0; hh < H; ++hh) {
    const float v  = lrelu(as[s * H + hh] + ad[d * H + hh]);
    const float m  = funkey(smax[d * H + hh]);
    const float ex = __expf(v - m);
    eex[(size_t)e * H + hh] = ex;
    atomicAdd(&ssum[d * H + hh], ex);
  }
}

// ---------------------------------------------------------------------------
// Edge pass 3: agg[dst, f] += h[src, f] * (ex / (denom + 1e-16)).
// One wave32 per edge; lanes stride the H*C feature dimension (coalesced
// gather from L2-resident h, global_atomic_add_f32 scatter).
// ---------------------------------------------------------------------------
__global__ __launch_bounds__(256)
void edge_agg_kernel(const long long* __restrict__ ei,
                     const float* __restrict__ hmat,
                     const float* __restrict__ eex, const float* __restrict__ ssum,
                     float* __restrict__ agg, int H, int C)
{
  const int wid  = (int)((blockIdx.x * blockDim.x + threadIdx.x) >> 5);
  const int lane = threadIdx.x & 31;
  if (wid >= ETOT) return;
  int s, d; edge_nodes(ei, wid, s, d);
  const int HC = H * C;
  const float* __restrict__ hs = hmat + (size_t)s * HC;
  float* __restrict__ ag       = agg  + (size_t)d * HC;
  for (int f = lane; f < HC; f += 32) {
    const int   hh = f / C;
    const float w  = eex[(size_t)wid * H + hh] / (ssum[d * H + hh] + 1e-16f);
    atomicAdd(&ag[f], hs[f] * w);
  }
}

// ---------------------------------------------------------------------------
// Finalize: out = act(agg + bias). blockDim.x == HC, one block per node.
// Safe in-place (elementwise).
// ---------------------------------------------------------------------------
__global__ void bias_act_kernel(const float* __restrict__ agg,
                                const float* __restrict__ bias,
                                float* __restrict__ out, int act)
{
  const int i = blockIdx.x * blockDim.x + threadIdx.x;
  float v = agg[i] + bias[threadIdx.x];
  if (act == 1) v = fmaxf(v, 0.f);
  else if (act == 2) v = tanhf(v);
  out[i] = v;
}

// ---------------------------------------------------------------------------
// Host-side: one full GAT layer (alpha -> segment softmax -> aggregate).
// The layer GEMM is launched by the caller (templated per shape).
// ---------------------------------------------------------------------------
static void run_edge_phase(const float* h_buf, const float* a_s, const float* a_d,
                           const float* b, int H, int C, int act,
                           float* agg, float* asb, float* adb,
                           unsigned* smax, float* ssum, float* eex,
                           const long long* ei, float* xout, hipStream_t stream)
{
  const int HC = H * C;
  const int nh = N_NODES * H;

  alpha_kernel<<<(nh * 32 + 255) / 256, 256, 0, stream>>>(h_buf, a_s, a_d, asb, adb, H, C);

  fill_u32<<<(nh + 255) / 256, 256, 0, stream>>>(smax, KEY_NEG_INF, nh);
  fill_f32<<<(nh + 255) / 256, 256, 0, stream>>>(ssum, 0.f, nh);
  fill_f32<<<(N_NODES * HC + 255) / 256, 256, 0, stream>>>(agg, 0.f, N_NODES * HC);

  edge_max_kernel   <<<(ETOT + 255) / 256, 256, 0, stream>>>(ei, asb, adb, smax, H);
  edge_expsum_kernel<<<(ETOT + 255) / 256, 256, 0, stream>>>(ei, asb, adb, smax, ssum, eex, H);

  const long long agg_threads = (long long)ETOT * 32;
  edge_agg_kernel<<<(unsigned)((agg_threads + 255) / 256), 256, 0, stream>>>(
      ei, h_buf, eex, ssum, agg, H, C);

  bias_act_kernel<<<N_NODES, HC, 0, stream>>>(agg, b, xout, act);
}

// ---------------------------------------------------------------------------
extern "C" void kernel_launch(void* const* d_in, const int* in_sizes, int n_in,
                              void* d_out, int out_size, void* d_ws, size_t ws_size,
                              hipStream_t stream)
{
  (void)in_sizes; (void)n_in; (void)out_size; (void)ws_size;

  const float*     x   = (const float*)d_in[0];
  const long long* ei  = (const long long*)d_in[1];   // int64 edge_index [2, E]
  const float*     W1  = (const float*)d_in[2];
  const float*     a1s = (const float*)d_in[3];
  const float*     a1d = (const float*)d_in[4];
  const float*     b1  = (const float*)d_in[5];
  const float*     W2  = (const float*)d_in[6];
  const float*     a2s = (const float*)d_in[7];
  const float*     a2d = (const float*)d_in[8];
  const float*     b2  = (const float*)d_in[9];
  const float*     W3  = (const float*)d_in[10];
  const float*     a3s = (const float*)d_in[11];
  const float*     a3d = (const float*)d_in[12];
  const float*     b3  = (const float*)d_in[13];
  const float*     Wc  = (const float*)d_in[14];
  const float*     bc  = (const float*)d_in[15];
  float* out = (float*)d_out;

  // ---- workspace carve (~170 MB) ----
  char* ws = (char*)d_ws;
  float*    h_buf = (float*)ws;    ws += sizeof(float)    * (size_t)N_NODES * 384;
  float*    agg   = (float*)ws;    ws += sizeof(float)    * (size_t)N_NODES * 384;
  float*    asb   = (float*)ws;    ws += sizeof(float)    * (size_t)N_NODES * 4;
  float*    adb   = (float*)ws;    ws += sizeof(float)    * (size_t)N_NODES * 4;
  unsigned* smax  = (unsigned*)ws; ws += sizeof(unsigned) * (size_t)N_NODES * 4;
  float*    ssum  = (float*)ws;    ws += sizeof(float)    * (size_t)N_NODES * 4;
  float*    eex   = (float*)ws;    ws += sizeof(float)    * (size_t)ETOT * 4;

  const dim3 gblock(128);
  const int  mtiles = N_NODES / 16;   // 3125

  // ---- Layer 1: [N,256] @ [256,384] ; H=4, C=96, ReLU ----
  gemm_wmma_f32<256, 384, false, 0><<<dim3(384 / 64, mtiles), gblock, 0, stream>>>(
      x, W1, nullptr, h_buf);
  run_edge_phase(h_buf, a1s, a1d, b1, 4, 96, /*relu*/1,
                 agg, asb, adb, smax, ssum, eex, ei, agg, stream);

  // ---- Layer 2: [N,384] @ [384,128] ; H=1, C=128, tanh ----
  gemm_wmma_f32<384, 128, false, 0><<<dim3(128 / 64, mtiles), gblock, 0, stream>>>(
      agg, W2, nullptr, h_buf);
  run_edge_phase(h_buf, a2s, a2d, b2, 1, 128, /*tanh*/2,
                 agg, asb, adb, smax, ssum, eex, ei, agg, stream);

  // ---- Layer 3: [N,128] @ [128,64] ; H=1, C=64, tanh ----
  gemm_wmma_f32<128, 64, false, 0><<<dim3(1, mtiles), gblock, 0, stream>>>(
      agg, W3, nullptr, h_buf);
  run_edge_phase(h_buf, a3s, a3d, b3, 1, 64, /*tanh*/2,
                 agg, asb, adb, smax, ssum, eex, ei, agg, stream);

  // ---- Classifier: [N,64] @ [64,40] + bc -> d_out (column-guarded) ----
  gemm_wmma_f32<64, 40, true, 0><<<dim3(1, mtiles), gblock, 0, stream>>>(
      agg, Wc, bc, out);
}